// LUTFR_5239860101663
// MI455X (gfx1250) — compile-verified
//
#include <hip/hip_runtime.h>
#include <math.h>

typedef float v2f __attribute__((ext_vector_type(2)));
typedef float v8f __attribute__((ext_vector_type(8)));

#define LUT_D  33
#define LUT_N  8
#define D3     (LUT_D * LUT_D * LUT_D)   /* 35937 */
#define IMG_H  2160
#define IMG_W  3840
#define HW     (IMG_H * IMG_W)           /* 8,294,400 */

// ---------------------------------------------------------------------------
// Kernel 1: softmax(lc) + einsum('ncdef,n->cdef') via V_WMMA_F32_16X16X4_F32.
// Output layout per stage: [D3][4] floats, channels interleaved (w unused),
// so the pixel kernel can fetch a corner pair with two B128 loads.
//
// Per wave: 16 consecutive spatial entries m, one channel c, one stage.
//   D[i][j] = sum_k A[i][k] * B[k][j],  A[i][k] = lut[stage,k(+4),c,m_base+i],
//   B[k][j] = w[k] (constant along N) -> every column of D holds the result.
// Two chained WMMAs cover K = 0..7. f32 in, f32 accumulate: exact.
// ---------------------------------------------------------------------------
__global__ __launch_bounds__(256) void lutfr_build_cluts(
    const float* __restrict__ lut,    // [2, 8, 3, 33, 33, 33]
    const float* __restrict__ lc0,    // [8]
    const float* __restrict__ lc1,    // [8]
    float* __restrict__ ws)           // [2][D3*4]
{
  const int CHUNKS = (D3 + 15) / 16;                       // 2247
  const int wave = (int)((blockIdx.x * blockDim.x + threadIdx.x) >> 5);
  const int lane = (int)(threadIdx.x & 31);
  if (wave >= 2 * 3 * CHUNKS) return;                      // whole wave exits

  const int stage = wave / (3 * CHUNKS);
  const int rem   = wave % (3 * CHUNKS);
  const int c     = rem / CHUNKS;
  const int mbase = (rem % CHUNKS) * 16;

  // softmax over 8 logits (redundant per wave; trivial, all L2/K$ hits)
  const float* lc = stage ? lc1 : lc0;
  float e[LUT_N];
  float mx = lc[0];
  #pragma unroll
  for (int n = 1; n < LUT_N; ++n) mx = fmaxf(mx, lc[n]);
  float s = 0.f;
  #pragma unroll
  for (int n = 0; n < LUT_N; ++n) { e[n] = __expf(lc[n] - mx); s += e[n]; }
  const float inv = 1.f / s;

  // A-matrix 16x4 f32 layout: VGPR0 = K={0 | 2}, VGPR1 = K={1 | 3}
  const int row = lane & 15;                 // M index
  const int kh  = lane >> 4;                 // 0 -> K 0,1 ; 1 -> K 2,3
  int m = mbase + row;
  if (m >= D3) m = D3 - 1;                   // clamp tail loads in-bounds

  const float* Lb = lut + (size_t)(stage * LUT_N * 3 + c) * D3;  // + n*3*D3 + m
  const size_t nstride = (size_t)3 * D3;

  v2f a0, a1, b0, b1;
  a0.x = Lb[(size_t)(2 * kh + 0) * nstride + m];
  a0.y = Lb[(size_t)(2 * kh + 1) * nstride + m];
  a1.x = Lb[(size_t)(2 * kh + 4) * nstride + m];
  a1.y = Lb[(size_t)(2 * kh + 5) * nstride + m];
  b0.x = e[2 * kh + 0] * inv;  b0.y = e[2 * kh + 1] * inv;
  b1.x = e[2 * kh + 4] * inv;  b1.y = e[2 * kh + 5] * inv;

  v8f acc = {};
  acc = __builtin_amdgcn_wmma_f32_16x16x4_f32(false, a0, false, b0,
                                              (short)0, acc, false, false);
  acc = __builtin_amdgcn_wmma_f32_16x16x4_f32(false, a1, false, b1,
                                              (short)0, acc, false, false);

  // D layout: VGPR v, lanes 0-15 -> (M=v, N=lane), lanes 16-31 -> (M=v+8).
  // All N columns are identical; lanes with N==0 (lane 0, lane 16) write out.
  if (row == 0) {
    float* outp = ws + (size_t)stage * (D3 * 4);
    #pragma unroll
    for (int v = 0; v < 8; ++v) {
      const int mo = mbase + v + 8 * kh;
      if (mo < D3) outp[(size_t)mo * 4 + c] = acc[v];
    }
  }
}

// ---------------------------------------------------------------------------
// Trilinear fetch from a [33,33,33,4] channel-interleaved LUT.
// Corner pair (r, r+1) = two GLOBAL_LOAD_B128s; 8 B128s per pixel per stage.
// ---------------------------------------------------------------------------
__device__ __forceinline__ void lutfr_tri(const float* __restrict__ clut,
                                          float r, float g, float b,
                                          float& o0, float& o1, float& o2)
{
  const float xr = fminf(fmaxf(r, 0.f), 1.f) * (float)(LUT_D - 1);
  const float xg = fminf(fmaxf(g, 0.f), 1.f) * (float)(LUT_D - 1);
  const float xb = fminf(fmaxf(b, 0.f), 1.f) * (float)(LUT_D - 1);
  const int ir = min((int)xr, LUT_D - 2);
  const int ig = min((int)xg, LUT_D - 2);
  const int ib = min((int)xb, LUT_D - 2);
  const float fr = xr - (float)ir;
  const float fg = xg - (float)ig;
  const float fb = xb - (float)ib;

  o0 = 0.f; o1 = 0.f; o2 = 0.f;
  #pragma unroll
  for (int dk = 0; dk < 2; ++dk) {
    const float wb = dk ? fb : 1.f - fb;
    #pragma unroll
    for (int dj = 0; dj < 2; ++dj) {
      const float wg = dj ? fg : 1.f - fg;
      const int idx = ((ib + dk) * LUT_D + (ig + dj)) * LUT_D + ir;
      const float4* ent = (const float4*)(clut + (size_t)idx * 4);
      const float4 c0 = ent[0];            // corner di=0: channels 0..2
      const float4 c1 = ent[1];            // corner di=1
      const float w0 = wb * wg * (1.f - fr);
      const float w1 = wb * wg * fr;
      o0 += w0 * c0.x + w1 * c1.x;
      o1 += w0 * c0.y + w1 * c1.y;
      o2 += w0 * c0.z + w1 * c1.z;
    }
  }
}

// ---------------------------------------------------------------------------
// Kernel 2: fused double LUT apply. 4 pixels/thread, planar B128 in/out.
// Intermediate image stays in registers (saves ~200 MB of HBM traffic).
// ---------------------------------------------------------------------------
__global__ __launch_bounds__(256) void lutfr_apply(
    const float* __restrict__ gt,    // [3, H, W]
    const float* __restrict__ ws,    // clut0 | clut1, each [D3*4]
    float* __restrict__ out)         // [3, H, W]
{
  const int t = (int)(blockIdx.x * blockDim.x + threadIdx.x);
  if (t >= HW / 4) return;

  const float4 R = ((const float4*)(gt + 0 * (size_t)HW))[t];
  const float4 G = ((const float4*)(gt + 1 * (size_t)HW))[t];
  const float4 B = ((const float4*)(gt + 2 * (size_t)HW))[t];

  const float* clut0 = ws;
  const float* clut1 = ws + (size_t)D3 * 4;

  float r[4] = {R.x, R.y, R.z, R.w};
  float g[4] = {G.x, G.y, G.z, G.w};
  float b[4] = {B.x, B.y, B.z, B.w};
  float orr[4], og[4], ob[4];

  #pragma unroll
  for (int k = 0; k < 4; ++k) {
    float sr, sg, sb;
    lutfr_tri(clut0, r[k], g[k], b[k], sr, sg, sb);        // stage 0 -> I_s
    lutfr_tri(clut1, sr, sg, sb, orr[k], og[k], ob[k]);    // stage 1 -> I_f
  }

  float4 OR = {orr[0], orr[1], orr[2], orr[3]};
  float4 OG = {og[0],  og[1],  og[2],  og[3]};
  float4 OB = {ob[0],  ob[1],  ob[2],  ob[3]};
  ((float4*)(out + 0 * (size_t)HW))[t] = OR;
  ((float4*)(out + 1 * (size_t)HW))[t] = OG;
  ((float4*)(out + 2 * (size_t)HW))[t] = OB;
}

// ---------------------------------------------------------------------------
extern "C" void kernel_launch(void* const* d_in, const int* in_sizes, int n_in,
                              void* d_out, int out_size, void* d_ws, size_t ws_size,
                              hipStream_t stream)
{
  const float* gt  = (const float*)d_in[0];   // [3,2160,3840]
  const float* lut = (const float*)d_in[1];   // [2,8,3,33,33,33]
  const float* lc0 = (const float*)d_in[2];   // [8]
  const float* lc1 = (const float*)d_in[3];   // [8]
  float* ws  = (float*)d_ws;                  // needs 2*D3*4*4 = ~1.15 MB
  float* out = (float*)d_out;

  const int CHUNKS  = (D3 + 15) / 16;         // 2247
  const int waves   = 2 * 3 * CHUNKS;         // 13482
  const int blocks1 = (waves * 32 + 255) / 256;
  lutfr_build_cluts<<<blocks1, 256, 0, stream>>>(lut, lc0, lc1, ws);

  const int blocks2 = (HW / 4 + 255) / 256;   // 8100
  lutfr_apply<<<blocks2, 256, 0, stream>>>(gt, ws, out);
}